// mHC_2989297238725
// MI455X (gfx1250) — compile-verified
//
#include <hip/hip_runtime.h>
#include <math.h>
#include <stdint.h>

typedef __attribute__((ext_vector_type(2))) float v2f;
typedef __attribute__((ext_vector_type(8))) float v8f;

#define MROWS 8192
#define KDIM  8192
#define DDIM  2048
#define SINK_ITERS 20
#define SINK_EPS 1e-20f
#define RMS_EPS  1e-9f

#define KSPAN  (KDIM / 4)   // K range per wave (4 waves split K)
#define CHUNK  64           // phi K-rows staged per async copy (64*32*4B = 8KB)
#define NCHUNK (KSPAN / CHUNK)

// one async b128 copy: IOFFSET applies to BOTH the global and LDS address
#define ASYNC_CP(OFF) \
  asm volatile("global_load_async_to_lds_b128 %0, %1, off offset:" OFF \
               :: "v"(la), "v"(ga) : "memory")

// ---------------------------------------------------------------------------
// Kernel 1: H = x @ phi via fp32 WMMA 16x16x4, with phi staged global->LDS by
// the async-DMA path (ASYNCcnt) and double-buffered per wave. Also computes
// inv_rms, sinkhorn, sigmoids; emits h_pre/h_res/h_post and the fused mix
// matrix W = h_res + h_post (x) h_pre.
// ---------------------------------------------------------------------------
__global__ __launch_bounds__(128) void mhc_stats_kernel(
    const float* __restrict__ x, const float* __restrict__ phi,
    const float* __restrict__ ab,
    float* __restrict__ h_pre, float* __restrict__ h_res,
    float* __restrict__ h_post, float* __restrict__ Wmix)
{
  __shared__ float phis[4][2][CHUNK * 32]; // 64KB: per-wave double buffer
  __shared__ float Hpart[4][16][32];       // per-wave partial H tiles
  __shared__ float sspart[4][16];          // per-wave partial sum-squares
  __shared__ float Hs[16][32];             // reduced H tile
  __shared__ float rms_s[16];              // inv_rms per row

  const int tid  = threadIdx.x;
  const int wave = tid >> 5;
  const int lane = tid & 31;
  const int r    = lane & 15;          // row within tile / N within tile
  const int kg   = (lane >> 4) << 1;   // K sub-offset: 0 (lanes 0-15) / 2 (16-31)
  const int row_base = blockIdx.x << 4;

  const float* xrow    = x   + (size_t)(row_base + r) * KDIM + (size_t)wave * KSPAN + kg;
  const float* phibase = phi + (size_t)wave * KSPAN * 32;

  v8f c0 = {};   // C tile, N = 0..15
  v8f c1 = {};   // C tile, N = 16..31
  float ss = 0.0f;

  // ---- async stage of phi chunk 0 into buffer 0 -------------------------
  {
    const uint32_t la = (uint32_t)(uintptr_t)&phis[wave][0][0] + (uint32_t)lane * 16u;
    const uint64_t ga = (uint64_t)(uintptr_t)phibase + (uint64_t)lane * 16u;
    ASYNC_CP("0");    ASYNC_CP("512");  ASYNC_CP("1024"); ASYNC_CP("1536");
    ASYNC_CP("2048"); ASYNC_CP("2560"); ASYNC_CP("3072"); ASYNC_CP("3584");
    ASYNC_CP("4096"); ASYNC_CP("4608"); ASYNC_CP("5120"); ASYNC_CP("5632");
    ASYNC_CP("6144"); ASYNC_CP("6656"); ASYNC_CP("7168"); ASYNC_CP("7680");
  }
  asm volatile("s_wait_asynccnt 0x0" ::: "memory");

  for (int c = 0; c < NCHUNK; ++c) {
    const int cur = c & 1;
    // ---- prefetch chunk c+1 into the other buffer (overlaps the WMMAs) --
    if (c + 1 < NCHUNK) {
      const uint32_t la = (uint32_t)(uintptr_t)&phis[wave][cur ^ 1][0] + (uint32_t)lane * 16u;
      const uint64_t ga = (uint64_t)(uintptr_t)(phibase + (size_t)(c + 1) * CHUNK * 32)
                        + (uint64_t)lane * 16u;
      ASYNC_CP("0");    ASYNC_CP("512");  ASYNC_CP("1024"); ASYNC_CP("1536");
      ASYNC_CP("2048"); ASYNC_CP("2560"); ASYNC_CP("3072"); ASYNC_CP("3584");
      ASYNC_CP("4096"); ASYNC_CP("4608"); ASYNC_CP("5120"); ASYNC_CP("5632");
      ASYNC_CP("6144"); ASYNC_CP("6656"); ASYNC_CP("7168"); ASYNC_CP("7680");
    }

    // ---- consume chunk c: 16 WMMA k-steps fed from LDS ------------------
    const float* xc = xrow + (size_t)c * CHUNK;
    const float* lp = &phis[wave][cur][0];
    #pragma unroll 4
    for (int kk = 0; kk < CHUNK; kk += 4) {
      const v2f a = *(const v2f*)(xc + kk);        // global b64, contiguous
      const float* p = lp + (kk + kg) * 32;        // LDS reads (DS pipe)
      v2f b0, b1;
      b0.x = p[r];       b0.y = p[32 + r];         // N tile 0, K = kg, kg+1
      b1.x = p[16 + r];  b1.y = p[48 + r];         // N tile 1
      ss = fmaf(a.x, a.x, fmaf(a.y, a.y, ss));
      c0 = __builtin_amdgcn_wmma_f32_16x16x4_f32(false, a, false, b0, (short)0, c0, false, false);
      c1 = __builtin_amdgcn_wmma_f32_16x16x4_f32(false, a, false, b1, (short)0, c1, false, false);
    }

    // prefetch must be complete before next iteration consumes it
    asm volatile("s_wait_asynccnt 0x0" ::: "memory");
  }

  // combine sum-squares across the two half-wave K groups (wave32)
  ss += __shfl_xor(ss, 16);
  if (lane < 16) sspart[wave][lane] = ss;

  // C layout: VGPR g, lane L -> M = g + 8*(L>>4), N = L&15
  const int mo = (lane >> 4) << 3;
  #pragma unroll
  for (int g = 0; g < 8; ++g) {
    Hpart[wave][mo + g][r]      = c0[g];
    Hpart[wave][mo + g][16 + r] = c1[g];
  }
  __syncthreads();

  // reduce 4 partial tiles -> Hs
  const float* hq0 = &Hpart[0][0][0];
  const float* hq1 = &Hpart[1][0][0];
  const float* hq2 = &Hpart[2][0][0];
  const float* hq3 = &Hpart[3][0][0];
  float* hflat = &Hs[0][0];
  for (int i = tid; i < 512; i += 128)
    hflat[i] = hq0[i] + hq1[i] + hq2[i] + hq3[i];

  if (tid < 16) {
    float s = sspart[0][tid] + sspart[1][tid] + sspart[2][tid] + sspart[3][tid];
    rms_s[tid] = rsqrtf(s * (1.0f / (float)KDIM) + RMS_EPS);
  }
  __syncthreads();

  // per-row epilogue: sigmoids + 20-iteration 4x4 sinkhorn, all in registers
  if (tid < 16) {
    const int   row   = row_base + tid;
    const float inv   = rms_s[tid];
    const float a_pre = ab[24], a_res = ab[25], a_post = ab[26];

    float hp[4], hq[4], pm[16];
    #pragma unroll
    for (int j = 0; j < 4; ++j) {
      hp[j] = 1.0f / (1.0f + __expf(-(inv * a_pre  * Hs[tid][j]      + ab[j])));
      hq[j] = 2.0f / (1.0f + __expf(-(inv * a_post * Hs[tid][20 + j] + ab[20 + j])));
    }
    #pragma unroll
    for (int i = 0; i < 16; ++i)
      pm[i] = __expf(inv * a_res * Hs[tid][4 + i] + ab[4 + i]);

    for (int it = 0; it < SINK_ITERS; ++it) {
      #pragma unroll
      for (int t = 0; t < 4; ++t) {
        float s = pm[4*t] + pm[4*t+1] + pm[4*t+2] + pm[4*t+3] + SINK_EPS;
        float rinv = 1.0f / s;
        pm[4*t] *= rinv; pm[4*t+1] *= rinv; pm[4*t+2] *= rinv; pm[4*t+3] *= rinv;
      }
      #pragma unroll
      for (int sc = 0; sc < 4; ++sc) {
        float s = pm[sc] + pm[4+sc] + pm[8+sc] + pm[12+sc] + SINK_EPS;
        float cinv = 1.0f / s;
        pm[sc] *= cinv; pm[4+sc] *= cinv; pm[8+sc] *= cinv; pm[12+sc] *= cinv;
      }
    }

    #pragma unroll
    for (int j = 0; j < 4; ++j) {
      h_pre [row*4 + j] = hp[j];
      h_post[row*4 + j] = hq[j];
    }
    #pragma unroll
    for (int t = 0; t < 4; ++t)
      #pragma unroll
      for (int s2 = 0; s2 < 4; ++s2) {
        h_res[row*16 + 4*t + s2] = pm[4*t + s2];
        Wmix [row*16 + 4*t + s2] = fmaf(hq[t], hp[s2], pm[4*t + s2]);
      }
  }
}

// ---------------------------------------------------------------------------
// Kernel 2: out[m,t,d] = sum_s W[m,t,s] * x[m,s,d]   (pure streaming)
// One block per row; 256 threads x 8 columns; b128 loads/stores.
// ---------------------------------------------------------------------------
__global__ __launch_bounds__(256) void mhc_mix_kernel(
    const float* __restrict__ x, const float* __restrict__ Wmix,
    float* __restrict__ out)
{
  const int m = blockIdx.x;
  const float* wrow = Wmix + (size_t)m * 16;
  float W[16];
  #pragma unroll
  for (int i = 0; i < 16; ++i) W[i] = wrow[i];   // uniform -> scalar loads

  const float* xr   = x   + (size_t)m * KDIM;
  float*       orow = out + (size_t)m * KDIM;
  const int d0 = threadIdx.x << 3;

  #pragma unroll
  for (int h = 0; h < 2; ++h) {
    const int d = d0 + (h << 2);
    const float4 x0 = *(const float4*)(xr + 0*DDIM + d);
    const float4 x1 = *(const float4*)(xr + 1*DDIM + d);
    const float4 x2 = *(const float4*)(xr + 2*DDIM + d);
    const float4 x3 = *(const float4*)(xr + 3*DDIM + d);
    #pragma unroll
    for (int t = 0; t < 4; ++t) {
      const float w0 = W[4*t], w1 = W[4*t+1], w2 = W[4*t+2], w3 = W[4*t+3];
      float4 o;
      o.x = fmaf(w3, x3.x, fmaf(w2, x2.x, fmaf(w1, x1.x, w0 * x0.x)));
      o.y = fmaf(w3, x3.y, fmaf(w2, x2.y, fmaf(w1, x1.y, w0 * x0.y)));
      o.z = fmaf(w3, x3.z, fmaf(w2, x2.z, fmaf(w1, x1.z, w0 * x0.z)));
      o.w = fmaf(w3, x3.w, fmaf(w2, x2.w, fmaf(w1, x1.w, w0 * x0.w)));
      *(float4*)(orow + t*DDIM + d) = o;
    }
  }
}

extern "C" void kernel_launch(void* const* d_in, const int* in_sizes, int n_in,
                              void* d_out, int out_size, void* d_ws, size_t ws_size,
                              hipStream_t stream) {
  const float* x   = (const float*)d_in[0];
  const float* phi = (const float*)d_in[1];
  const float* ab  = (const float*)d_in[2];

  float* outp   = (float*)d_out;
  float* h_pre  = outp;                        // M*4
  float* h_res  = outp + (size_t)MROWS * 4;    // M*16
  float* h_post = outp + (size_t)MROWS * 20;   // M*4
  float* big    = outp + (size_t)MROWS * 24;   // M*8192
  float* Wmix   = (float*)d_ws;                // M*16 scratch

  mhc_stats_kernel<<<MROWS / 16, 128, 0, stream>>>(x, phi, ab, h_pre, h_res, h_post, Wmix);
  mhc_mix_kernel<<<MROWS, 256, 0, stream>>>(x, Wmix, big);
}